// CWT_28716151341384
// MI455X (gfx1250) — compile-verified
//
#include <hip/hip_runtime.h>
#include <math.h>

typedef __attribute__((ext_vector_type(2))) float v2f;
typedef __attribute__((ext_vector_type(8))) float v8f;

// Problem constants (baked into the reference)
constexpr int T_LEN   = 3000;
constexpr int NBATCH  = 16;
constexpr int NCH     = 64;
constexpr int NSCALE  = 50;
constexpr int NSIG    = NBATCH * NCH;          // 1024 signals
constexpr int JMAX    = 402;                   // signal stored at s_sig[JMAX + t]
constexpr int SIG_LDS = 3888;                  // covers max read t0+528 for shift-packed tile
constexpr int OUT_TSTRIDE = NCH * NSCALE;      // 3200 floats between consecutive t

// f-tiles 0..2: scales 16ft+1 .. 16ft+16, K-extent 16*amax+4
__device__ __host__ constexpr int tile_amax(int ft) { return 16 * (ft + 1); }
__device__ __host__ constexpr int tile_kext(int ft) { return 16 * tile_amax(ft) + 4; }
__device__ __host__ constexpr int tile_foff(int ft) {
    return ft == 0 ? 0 : ft == 1 ? 4160 : ft == 2 ? 12416 : 24768;
}
// f-tile 3 (scales 49,50): time-shift packed, column n = scale 49+(n&1), shift 16*(n>>1)
constexpr int K3 = 916;                        // 402 + 112 + 400 + 2, multiple of 4
constexpr int FILT_TOTAL = 24768 + K3 * 16;    // 39424 floats

// ---------------------------------------------------------------------------
// Kernel 1: int_psi (pywt integrate_wavelet of 'morl', 1024 pts, float64)
// ---------------------------------------------------------------------------
__global__ void cwt_intpsi(double* __restrict__ int_psi) {
    if (blockIdx.x == 0 && threadIdx.x == 0) {
        const double step = 16.0 / 1023.0;
        double acc = 0.0;
        for (int i = 0; i < 1024; ++i) {
            double xg  = -8.0 + (double)i * step;
            double psi = exp(-0.5 * xg * xg) * cos(5.0 * xg);
            acc += psi;
            int_psi[i] = acc * step;
        }
    }
}

// H_a[j] for j in [-(8a+1), 8a]: diff-fused, -sqrt(a)-scaled 'morl' filter
__device__ inline float morl_H(const double* __restrict__ int_psi, int a, int j) {
    const int L = 16 * a + 1;
    const int r = j + 8 * a + 1;               // index into g (valid 0..L)
    if (r < 0 || r > L) return 0.0f;
    const double astep = (double)a * (16.0 / 1023.0);
    auto fval = [&](int rr) -> float {
        if (rr < 0 || rr >= L) return 0.0f;
        long idx = (long)((double)rr / astep); // numpy trunc-to-int64
        if (idx >= 1024) return 0.0f;
        return (float)int_psi[idx];
    };
    float g = fval(r - 1) - fval(r);           // diff folded into kernel
    return -sqrtf((float)a) * g;
}

// ---------------------------------------------------------------------------
// Kernel 2: filter bank slabs, H[k][16] layout per f-tile.
//   ft 0..2 : column nf = scale 16ft+nf+1, k = j + (8*amax+2)
//   ft 3    : column nf = scale 49+(nf&1) shifted by delta=16*(nf>>1);
//             k = j + 402 + delta  (K-extent 916)
// ---------------------------------------------------------------------------
__global__ void cwt_filters(const double* __restrict__ int_psi, float* __restrict__ filt) {
    const int ft = blockIdx.x;
    if (ft < 3) {
        const int amax = 16 * (ft + 1);
        const int J    = 8 * amax + 2;
        const int Kext = 2 * J;
        const int base = tile_foff(ft);
        for (int i = threadIdx.x; i < Kext * 16; i += blockDim.x) {
            const int jj = i >> 4;
            const int nf = i & 15;
            const int a  = 16 * ft + nf + 1;
            filt[base + i] = morl_H(int_psi, a, jj - J);
        }
    } else {
        const int base = tile_foff(3);
        for (int i = threadIdx.x; i < K3 * 16; i += blockDim.x) {
            const int jj    = i >> 4;
            const int nf    = i & 15;
            const int a     = 49 + (nf & 1);
            const int delta = 16 * (nf >> 1);
            filt[base + i] = morl_H(int_psi, a, jj - 402 - delta);
        }
    }
}

// ---------------------------------------------------------------------------
// Kernel 3: transpose x [N,T,C] -> xT [signal][t] contiguous
// ---------------------------------------------------------------------------
__global__ void cwt_transpose(const float* __restrict__ x, float* __restrict__ xT) {
    const size_t total = (size_t)NBATCH * T_LEN * NCH;
    for (size_t i = (size_t)blockIdx.x * blockDim.x + threadIdx.x; i < total;
         i += (size_t)gridDim.x * blockDim.x) {
        int c = (int)(i % NCH);
        size_t rest = i / NCH;
        int t = (int)(rest % T_LEN);
        int n = (int)(rest / T_LEN);
        xT[((size_t)(n * NCH + c)) * T_LEN + t] = x[i];
    }
}

// Store 8 rows of a D tile: acc[v] -> t = tbase + v (t-stride 3200 floats)
template <bool GUARD>
__device__ inline void store_tile(float* __restrict__ pbase, int tbase, v8f acc) {
    float* p = pbase + (size_t)tbase * OUT_TSTRIDE;
#pragma unroll
    for (int v = 0; v < 8; ++v) {
        if (!GUARD || (tbase + v) < T_LEN)
            p[(size_t)v * OUT_TSTRIDE] = acc[v];
    }
}

// Common LDS staging
__device__ inline void stage(const float* __restrict__ xT, const float* __restrict__ filt,
                             float* s_sig, float* s_filt, int si, int fbase, int kext, int tid) {
    for (int i = tid; i < SIG_LDS; i += 256) {
        int t = i - JMAX;
        s_sig[i] = (t >= 0 && t < T_LEN) ? xT[(size_t)si * T_LEN + t] : 0.0f;
    }
    for (int i = tid; i < kext * 16; i += 256) s_filt[i] = filt[fbase + i];
    __syncthreads();
}

// ---------------------------------------------------------------------------
// Kernel 4a: f-tiles 0..2 via V_WMMA_F32_16X16X4_F32. 1024 blocks x 8 waves;
// each wave computes 4 adjacent 16(t)x16(f) tiles per K-loop (4x B reuse).
// ---------------------------------------------------------------------------
template <int FT>
__global__ __launch_bounds__(256) void cwt_gemm(const float* __restrict__ xT,
                                                const float* __restrict__ filt,
                                                float* __restrict__ out) {
    constexpr int J      = 8 * tile_amax(FT) + 2;
    constexpr int Kext   = 2 * J;
    constexpr int NGROUP = 47;                 // 47 groups * 4 tiles = 188 t-tiles

    extern __shared__ float smem[];
    float* s_sig  = smem;
    float* s_filt = smem + SIG_LDS;

    const int si = blockIdx.x;
    const int n  = si >> 6;
    const int c  = si & 63;
    const int tid = threadIdx.x;
    stage(xT, filt, s_sig, s_filt, si, tile_foff(FT), Kext, tid);

    const int lane = tid & 31;
    const int wave = tid >> 5;
    const int half = lane >> 4;
    const int row  = lane & 15;
    const int f    = FT * 16 + row;

    float* pbase = out + (size_t)n * ((size_t)T_LEN * OUT_TSTRIDE) + c * NSCALE + f;

    for (int g = wave; g < NGROUP; g += 8) {
        const int t0 = g * 64;
        v8f acc[4];
#pragma unroll
        for (int q = 0; q < 4; ++q) acc[q] = v8f{};

        const float* aPtr = s_sig + (JMAX - J) + t0 + row + 2 * half;
        const float* bPtr = s_filt + 32 * half + row;

#pragma unroll 8
        for (int kk = 0; kk < Kext; kk += 4) {
            v2f b;
            b.x = bPtr[kk * 16];
            b.y = bPtr[kk * 16 + 16];
            v2f a[4];
#pragma unroll
            for (int q = 0; q < 4; ++q) {
                a[q].x = aPtr[kk + 16 * q];
                a[q].y = aPtr[kk + 16 * q + 1];
            }
#pragma unroll
            for (int q = 0; q < 4; ++q)
                acc[q] = __builtin_amdgcn_wmma_f32_16x16x4_f32(
                    false, a[q], false, b, (short)0, acc[q], false, false);
        }

#pragma unroll
        for (int q = 0; q < 4; ++q) {
            const int t0q   = t0 + q * 16;
            const int tbase = t0q + 8 * half;
            if (t0q + 16 <= T_LEN) store_tile<false>(pbase, tbase, acc[q]);
            else                   store_tile<true >(pbase, tbase, acc[q]);
        }
    }
}

// ---------------------------------------------------------------------------
// Kernel 4b: f-tile 3 (scales 49,50), time-shift packed: one 16x16 tile covers
// 128 t-values x 2 scales (all 16 B columns useful). 24 t-tiles of K=916;
// each of the 8 waves runs 3 tiles in one K-loop (3x B reuse).
// ---------------------------------------------------------------------------
__global__ __launch_bounds__(256) void cwt_gemm3(const float* __restrict__ xT,
                                                 const float* __restrict__ filt,
                                                 float* __restrict__ out) {
    extern __shared__ float smem[];
    float* s_sig  = smem;
    float* s_filt = smem + SIG_LDS;

    const int si = blockIdx.x;
    const int n  = si >> 6;
    const int c  = si & 63;
    const int tid = threadIdx.x;
    stage(xT, filt, s_sig, s_filt, si, tile_foff(3), K3, tid);

    const int lane = tid & 31;
    const int wave = tid >> 5;
    const int half = lane >> 4;
    const int row  = lane & 15;
    const int f     = 48 + (row & 1);          // packed column -> scale
    const int tcol  = 16 * (row >> 1);         // packed column -> time shift

    float* pbase = out + (size_t)n * ((size_t)T_LEN * OUT_TSTRIDE) + c * NSCALE + f;

    // Wave w owns tiles t0 = w*384 + {0,128,256}  (8 waves x 3 = 24 tiles)
    const int tw = wave * 384;
    v8f acc[3];
#pragma unroll
    for (int q = 0; q < 3; ++q) acc[q] = v8f{};

    // A[m][k] = s[t0 + m + k - 402] -> LDS index t0 + m + k (JMAX cancels)
    const float* aPtr = s_sig + tw + row + 2 * half;
    const float* bPtr = s_filt + 32 * half + row;

#pragma unroll 8
    for (int kk = 0; kk < K3; kk += 4) {
        v2f b;
        b.x = bPtr[kk * 16];
        b.y = bPtr[kk * 16 + 16];
        v2f a[3];
#pragma unroll
        for (int q = 0; q < 3; ++q) {
            a[q].x = aPtr[kk + 128 * q];
            a[q].y = aPtr[kk + 128 * q + 1];
        }
#pragma unroll
        for (int q = 0; q < 3; ++q)
            acc[q] = __builtin_amdgcn_wmma_f32_16x16x4_f32(
                false, a[q], false, b, (short)0, acc[q], false, false);
    }

#pragma unroll
    for (int q = 0; q < 3; ++q) {
        const int t0    = tw + q * 128;
        const int tbase = t0 + tcol + 8 * half;   // + v inside store_tile
        if (t0 + 128 <= T_LEN) store_tile<false>(pbase, tbase, acc[q]);
        else                   store_tile<true >(pbase, tbase, acc[q]);
    }
}

// ---------------------------------------------------------------------------
// Kernel 5: second tuple output y (int64 labels) appended as floats
// ---------------------------------------------------------------------------
__global__ void cwt_copy_y(const long long* __restrict__ y, float* __restrict__ out_tail) {
    int i = threadIdx.x;
    if (i < NBATCH) out_tail[i] = (float)y[i];
}

extern "C" void kernel_launch(void* const* d_in, const int* in_sizes, int n_in,
                              void* d_out, int out_size, void* d_ws, size_t ws_size,
                              hipStream_t stream) {
    const float*     x = (const float*)d_in[0];
    const long long* y = (const long long*)d_in[1];
    float* out = (float*)d_out;

    // Workspace layout
    char* ws = (char*)d_ws;
    double* int_psi = (double*)ws;                            // 1024 * 8 = 8192 B
    float*  filt    = (float*)(ws + 8192);                    // 39424 * 4 = 157696 B
    float*  xT      = (float*)(ws + 8192 + 157696);           // 3,072,000 * 4 B

    cwt_intpsi   <<<1, 64, 0, stream>>>(int_psi);
    cwt_filters  <<<4, 256, 0, stream>>>(int_psi, filt);
    cwt_transpose<<<3000, 256, 0, stream>>>(x, xT);

    size_t lds0 = (size_t)(SIG_LDS + tile_kext(0) * 16) * sizeof(float);
    size_t lds1 = (size_t)(SIG_LDS + tile_kext(1) * 16) * sizeof(float);
    size_t lds2 = (size_t)(SIG_LDS + tile_kext(2) * 16) * sizeof(float);
    size_t lds3 = (size_t)(SIG_LDS + K3 * 16) * sizeof(float);
    cwt_gemm<0><<<NSIG, 256, lds0, stream>>>(xT, filt, out);
    cwt_gemm<1><<<NSIG, 256, lds1, stream>>>(xT, filt, out);
    cwt_gemm<2><<<NSIG, 256, lds2, stream>>>(xT, filt, out);
    cwt_gemm3  <<<NSIG, 256, lds3, stream>>>(xT, filt, out);

    const size_t cwt_elems = (size_t)NBATCH * T_LEN * NCH * NSCALE;      // 153,600,000
    if ((size_t)out_size >= cwt_elems + NBATCH)
        cwt_copy_y<<<1, 32, 0, stream>>>(y, out + cwt_elems);
}